// SHIG_Model_63763084477188
// MI455X (gfx1250) — compile-verified
//
#include <hip/hip_runtime.h>
#include <hip/hip_bf16.h>

// SHIG signed-graph GNN for MI455X (gfx1250, wave32).
// Aggregations: fp32 scatter-atomics (L2-resident, bandwidth bound).
// GEMMs: V_WMMA_F32_16X16X4_F32, one wave per 16x64 output tile, fused
// bias + mean-scale + tanh epilogue.
//
// Workspace layout (floats):
//   [0,        N)        inv_p  (1/max(pos_deg,1))
//   [N,       2N)        inv_n
//   [2N, 2N+2*N*128)     agg buffers (layer0: 2 x N*128 ; layers: 4 x N*64)
// Total ~103.2 MB for N=100000.

typedef __attribute__((ext_vector_type(2))) float v2f;
typedef __attribute__((ext_vector_type(8))) float v8f;

__global__ void k_zero4(float4* __restrict__ p, long n4) {
  long t = (long)blockIdx.x * blockDim.x + threadIdx.x;
  if (t < n4) p[t] = make_float4(0.f, 0.f, 0.f, 0.f);
}

// count destinations: cnt[dst] += 1
__global__ void k_count(const long long* __restrict__ eidx, long E,
                        float* __restrict__ cnt) {
  long t = (long)blockIdx.x * blockDim.x + threadIdx.x;
  if (t < E) {
    int dst = (int)eidx[E + t];
    atomicAdd(cnt + dst, 1.0f);
  }
}

__global__ void k_invert(float* __restrict__ c, int n) {
  int t = blockIdx.x * blockDim.x + threadIdx.x;
  if (t < n) c[t] = 1.0f / fmaxf(c[t], 1.0f);
}

// raw segment-sum: agg[dst, :F] += X[src, :F]; one thread per (edge, float4)
// cshift = log2(F/4)
__global__ void k_scatter(const float* __restrict__ X, int xstride,
                          const long long* __restrict__ eidx, long E,
                          float* __restrict__ agg, int F, int cshift) {
  long t = (long)blockIdx.x * blockDim.x + threadIdx.x;
  long total = E << cshift;
  if (t >= total) return;
  long e = t >> cshift;
  int c = (int)(t & ((1 << cshift) - 1));
  int src = (int)eidx[e];
  int dst = (int)eidx[E + e];
  float4 v = *(const float4*)(X + (size_t)src * xstride + c * 4);
  float* ad = agg + (size_t)dst * F + c * 4;
  atomicAdd(ad + 0, v.x);
  atomicAdd(ad + 1, v.y);
  atomicAdd(ad + 2, v.z);
  atomicAdd(ad + 3, v.w);
}

// Accumulate acc[0..3] (16x64 tile) += diag(scale) * A[m0:m0+16, 0:K] @ W[K x 64]
// A-frag (16x4 f32): lanes 0-15 -> K = k,k+1 ; lanes 16-31 -> K = k+2,k+3
// B-frag (4x16 f32): lanes 0-15 -> K = k,k+1 rows ; lanes 16-31 -> K = k+2,k+3
__device__ __forceinline__ void gemm_src(const float* __restrict__ A, int lda,
                                         const float* __restrict__ sc,
                                         const float* __restrict__ W, int K,
                                         int m0, int lid, int lg, v8f acc[4]) {
  if (A == nullptr || K <= 0) return;
  int row = m0 + lid;
  float s = (sc != nullptr) ? sc[row] : 1.0f;
  const float* arow = A + (size_t)row * lda + 2 * lg;
  for (int k = 0; k < K; k += 4) {
    v2f a = *(const v2f*)(arow + k);
    a.x *= s;
    a.y *= s;
    int kk = k + 2 * lg;
    const float* w0 = W + (size_t)kk * 64 + lid;
#pragma unroll
    for (int nt = 0; nt < 4; ++nt) {
      v2f b;
      b.x = w0[nt * 16];        // W[kk    ][nt*16+lid]
      b.y = w0[64 + nt * 16];   // W[kk + 1][nt*16+lid]
      acc[nt] = __builtin_amdgcn_wmma_f32_16x16x4_f32(
          false, a, false, b, (short)0, acc[nt], false, false);
    }
  }
}

// out[m0:m0+16, 0:64] = tanh( sum_i scale_i*A_i@W_i + bias ), ldo-strided store.
__global__ void __launch_bounds__(128, 1) k_gemm_tanh(
    const float* A0, int lda0, const float* sc0, const float* W0, int K0,
    const float* A1, int lda1, const float* sc1, const float* W1, int K1,
    const float* A2, int lda2, const float* sc2, const float* W2, int K2,
    const float* __restrict__ bias, float* __restrict__ out, int ldo,
    int row_tiles) {
  int wv = blockIdx.x * (blockDim.x >> 5) + (threadIdx.x >> 5);
  if (wv >= row_tiles) return;  // wave-uniform: EXEC stays all-1 for WMMA
  int lane = threadIdx.x & 31;
  int lg = lane >> 4;
  int lid = lane & 15;
  int m0 = wv * 16;

  v8f acc[4];
#pragma unroll
  for (int nt = 0; nt < 4; ++nt) {
    float b = bias[nt * 16 + lid];
#pragma unroll
    for (int r = 0; r < 8; ++r) acc[nt][r] = b;
  }

  gemm_src(A0, lda0, sc0, W0, K0, m0, lid, lg, acc);
  gemm_src(A1, lda1, sc1, W1, K1, m0, lid, lg, acc);
  gemm_src(A2, lda2, sc2, W2, K2, m0, lid, lg, acc);

#pragma unroll
  for (int nt = 0; nt < 4; ++nt) {
#pragma unroll
    for (int r = 0; r < 8; ++r) {
      int row = m0 + r + 8 * lg;   // C/D layout: VGPR r -> M = r (+8 for hi lanes)
      out[(size_t)row * ldo + nt * 16 + lid] = tanhf(acc[nt][r]);
    }
  }
}

static inline unsigned nblk(long n, int bs) { return (unsigned)((n + bs - 1) / bs); }

extern "C" void kernel_launch(void* const* d_in, const int* in_sizes, int n_in,
                              void* d_out, int out_size, void* d_ws, size_t ws_size,
                              hipStream_t stream) {
  const float* x          = (const float*)d_in[0];
  const long long* pe     = (const long long*)d_in[1];
  const long long* ne     = (const long long*)d_in[2];
  const float* Wp_l       = (const float*)d_in[3];
  const float* Wp_r       = (const float*)d_in[4];
  const float* bp         = (const float*)d_in[5];
  const float* Wn_l       = (const float*)d_in[6];
  const float* Wn_r       = (const float*)d_in[7];
  const float* bn         = (const float*)d_in[8];
  const float* Wl_pos     = (const float*)d_in[9];
  const float* Wr_pos     = (const float*)d_in[10];
  const float* b_pos      = (const float*)d_in[11];
  const float* Wl_neg     = (const float*)d_in[12];
  const float* Wr_neg     = (const float*)d_in[13];
  const float* b_neg      = (const float*)d_in[14];

  const int N = in_sizes[0] / 128;       // 100000
  const long E = in_sizes[1] / 2;        // 1200000
  const int L1 = in_sizes[11] / 64;      // L-1 = 2

  float* ws    = (float*)d_ws;
  float* inv_p = ws;
  float* inv_n = ws + N;
  float* buf   = ws + 2 * (size_t)N;

  // --- degree counts -> reciprocals ---
  k_zero4<<<nblk((long)2 * N / 4, 256), 256, 0, stream>>>((float4*)inv_p, (long)2 * N / 4);
  k_count<<<nblk(E, 256), 256, 0, stream>>>(pe, E, inv_p);
  k_count<<<nblk(E, 256), 256, 0, stream>>>(ne, E, inv_n);
  k_invert<<<nblk((long)2 * N, 256), 256, 0, stream>>>(inv_p, 2 * N);

  const int row_tiles = N / 16;                 // 6250 (exact)
  const unsigned gblk = (row_tiles + 3) / 4;    // 4 waves / block
  const long bufN4 = (long)N * 256 / 4;         // 2*N*128 floats as float4
  float* z = (float*)d_out;

  // --- layer 0 ---
  k_zero4<<<nblk(bufN4, 256), 256, 0, stream>>>((float4*)buf, bufN4);
  float* agg_p = buf;
  float* agg_n = buf + (size_t)N * 128;
  k_scatter<<<nblk(E * 32, 256), 256, 0, stream>>>(x, 128, pe, E, agg_p, 128, 5);
  k_scatter<<<nblk(E * 32, 256), 256, 0, stream>>>(x, 128, ne, E, agg_n, 128, 5);

  k_gemm_tanh<<<gblk, 128, 0, stream>>>(agg_p, 128, inv_p, Wp_l, 128,
                                        x,     128, nullptr, Wp_r, 128,
                                        nullptr, 0, nullptr, nullptr, 0,
                                        bp, z, 128, row_tiles);
  k_gemm_tanh<<<gblk, 128, 0, stream>>>(agg_n, 128, inv_n, Wn_l, 128,
                                        x,     128, nullptr, Wn_r, 128,
                                        nullptr, 0, nullptr, nullptr, 0,
                                        bn, z + 64, 128, row_tiles);

  // --- layers 1..L-1 ---
  for (int l = 0; l < L1; ++l) {
    k_zero4<<<nblk(bufN4, 256), 256, 0, stream>>>((float4*)buf, bufN4);
    float* ap_p = buf;                        // pos-agg of x_p
    float* an_n = buf + (size_t)N * 64;       // neg-agg of x_n
    float* ap_n = buf + (size_t)2 * N * 64;   // pos-agg of x_n
    float* an_p = buf + (size_t)3 * N * 64;   // neg-agg of x_p
    // x_p = z[:, :64], x_n = z[:, 64:]  (stride 128)
    k_scatter<<<nblk(E * 16, 256), 256, 0, stream>>>(z,      128, pe, E, ap_p, 64, 4);
    k_scatter<<<nblk(E * 16, 256), 256, 0, stream>>>(z + 64, 128, ne, E, an_n, 64, 4);
    k_scatter<<<nblk(E * 16, 256), 256, 0, stream>>>(z + 64, 128, pe, E, ap_n, 64, 4);
    k_scatter<<<nblk(E * 16, 256), 256, 0, stream>>>(z,      128, ne, E, an_p, 64, 4);

    const float* Wlp = Wl_pos + (size_t)l * 128 * 64;
    const float* Wrp = Wr_pos + (size_t)l * 64 * 64;
    const float* blp = b_pos + (size_t)l * 64;
    const float* Wln = Wl_neg + (size_t)l * 128 * 64;
    const float* Wrn = Wr_neg + (size_t)l * 64 * 64;
    const float* bln = b_neg + (size_t)l * 64;

    // o_p = [ap_p | an_n] @ Wl_pos + x_p @ Wr_pos + b_pos  -> z[:, :64]
    k_gemm_tanh<<<gblk, 128, 0, stream>>>(ap_p, 64, inv_p, Wlp,           64,
                                          an_n, 64, inv_n, Wlp + 64 * 64, 64,
                                          z,   128, nullptr, Wrp,         64,
                                          blp, z, 128, row_tiles);
    // o_n = [ap_n | an_p] @ Wl_neg + x_n @ Wr_neg + b_neg  -> z[:, 64:]
    k_gemm_tanh<<<gblk, 128, 0, stream>>>(ap_n, 64, inv_p, Wln,           64,
                                          an_p, 64, inv_n, Wln + 64 * 64, 64,
                                          z + 64, 128, nullptr, Wrn,      64,
                                          bln, z + 64, 128, row_tiles);
  }
}